// SoftGraphConditionedInterventionalPFN_15719580303535
// MI455X (gfx1250) — compile-verified
//
#include <hip/hip_runtime.h>
#include <hip/hip_bf16.h>
#include <math.h>

typedef float v2f __attribute__((ext_vector_type(2)));
typedef float v8f __attribute__((ext_vector_type(8)));

#define B_   8
#define S_   512
#define F_   32
#define D_   256
#define H_   4
#define DH_  64

// ---------------------------------------------------------------------------
// CDNA5 async VMEM->LDS helpers (ASYNCcnt-tracked, no VGPR staging).
// lds_off = byte offset within the workgroup LDS allocation (low 32 bits of a
// generic pointer to a __shared__ object). Each enabled lane moves 16 bytes.
// ---------------------------------------------------------------------------
__device__ __forceinline__ void async_ld128(uint32_t lds_off, const void* gptr)
{
    asm volatile("global_load_async_to_lds_b128 %0, %1, off"
                 :: "v"(lds_off), "v"((unsigned long long)(uintptr_t)gptr)
                 : "memory");
}
__device__ __forceinline__ void async_wait0()
{
    asm volatile("s_wait_asynccnt 0x0" ::: "memory");
}
__device__ __forceinline__ uint32_t lds_off32(const void* p)
{
    return (uint32_t)(uintptr_t)p;   // LDS aperture: addr[31:0] is the LDS offset
}

// ---------------------------------------------------------------------------
// LayerNorm over last dim (D_=256). mode 0: contiguous rows.
// mode 1: gather from x1 layout (b, s, f, d) into (b, f, s') contiguous rows,
//         s = s0 + s' (train: s0=0, test: s0=256), 256 s' per (b,f).
// ---------------------------------------------------------------------------
__global__ __launch_bounds__(256)
void ln_rows(const float* __restrict__ in, float* __restrict__ out,
             const float* __restrict__ g, const float* __restrict__ bta,
             int mode, int s0)
{
    int r = blockIdx.x;
    int tid = threadIdx.x;
    size_t in_row;
    if (mode == 0) {
        in_row = (size_t)r;
    } else {
        int b   = r >> 13;        // / (F_*256)
        int rem = r & 8191;
        int f   = rem >> 8;
        int sp  = rem & 255;
        in_row = ((size_t)b * S_ + (size_t)(s0 + sp)) * F_ + f;
    }
    float x = in[in_row * D_ + tid];
    __shared__ float rs[256], rq[256];
    rs[tid] = x; rq[tid] = x * x;
    __syncthreads();
    for (int s = 128; s > 0; s >>= 1) {
        if (tid < s) { rs[tid] += rs[tid + s]; rq[tid] += rq[tid + s]; }
        __syncthreads();
    }
    float mean = rs[0] * (1.0f / D_);
    float var  = rq[0] * (1.0f / D_) - mean * mean;
    float inv  = rsqrtf(var + 1e-5f);
    out[(size_t)r * D_ + tid] = (x - mean) * inv * g[tid] + bta[tid];
}

// ---------------------------------------------------------------------------
// WMMA fp32 GEMM:  C[M,N] = A[M,K] @ W[N,K]^T + bias[N]   (+ epilogue)
// MODE 0: plain   MODE 1: exact GELU   MODE 2: += residual R (same layout)
// Block tile 128x64, BK=16, 8 waves (4x2), wave tile 32x32 (2x2 WMMA 16x16x4).
// Double-buffered LDS tiles filled with global_load_async_to_lds_b128.
// LDS row stride 20 floats (80B): 16B-aligned rows, conflict-free frag reads.
// Requires M%128==0, N%64==0, K%16==0 (always true here).
// ---------------------------------------------------------------------------
template<int MODE>
__global__ __launch_bounds__(256)
void gemm_wmma(const float* __restrict__ A, const float* __restrict__ W,
               const float* __restrict__ bias, const float* __restrict__ R,
               float* __restrict__ C, int M, int N, int K)
{
    const int BM = 128, BN = 64, BK = 16, LDT = 20;
    __shared__ __align__(16) float sA[2][BM * LDT];  // 2 x 10240 B
    __shared__ __align__(16) float sB[2][BN * LDT];  // 2 x  5120 B
    int tid  = threadIdx.x;
    int wave = tid >> 5;
    int lane = tid & 31;
    int half = lane >> 4;
    int idx  = lane & 15;
    int wr   = wave & 3;        // wave row 0..3  (32 rows each)
    int wc   = wave >> 2;       // wave col 0..1  (32 cols each)
    int m0 = blockIdx.y * BM;
    int n0 = blockIdx.x * BN;

    // Async fill of one BK tile into buffer bsel: A = 512 16B segments,
    // W = 256 16B segments; all 256 lanes issue async b128 transfers.
    auto fill = [&](int k0, int bsel) {
        #pragma unroll
        for (int rep = 0; rep < 2; rep++) {
            int seg = tid + rep * 256;          // 0..511
            int r = seg >> 2, q = (seg & 3) * 4;
            async_ld128(lds_off32(&sA[bsel][r * LDT + q]),
                        A + (size_t)(m0 + r) * K + k0 + q);
        }
        {
            int r = tid >> 2, q = (tid & 3) * 4;  // 0..255
            async_ld128(lds_off32(&sB[bsel][r * LDT + q]),
                        W + (size_t)(n0 + r) * K + k0 + q);
        }
    };

    v8f acc[2][2] = {};

    fill(0, 0);
    async_wait0();
    __syncthreads();

    int ntiles = K / BK;
    for (int t = 0; t < ntiles; t++) {
        int cur = t & 1;
        if (t + 1 < ntiles) fill((t + 1) * BK, cur ^ 1);

        const float* pa = sA[cur];
        const float* pb = sB[cur];
        #pragma unroll
        for (int kk = 0; kk < BK; kk += 4) {
            v2f af[2], bf[2];
            #pragma unroll
            for (int tm = 0; tm < 2; tm++) {
                const float* p = &pa[(wr * 32 + tm * 16 + idx) * LDT + kk + 2 * half];
                af[tm].x = p[0]; af[tm].y = p[1];     // A[m=idx][k=2*half+j]
            }
            #pragma unroll
            for (int tn = 0; tn < 2; tn++) {
                const float* p = &pb[(wc * 32 + tn * 16 + idx) * LDT + kk + 2 * half];
                bf[tn].x = p[0]; bf[tn].y = p[1];     // B[k=2*half+j][n=idx]
            }
            #pragma unroll
            for (int tm = 0; tm < 2; tm++)
                #pragma unroll
                for (int tn = 0; tn < 2; tn++)
                    acc[tm][tn] = __builtin_amdgcn_wmma_f32_16x16x4_f32(
                        false, af[tm], false, bf[tn], (short)0, acc[tm][tn], false, false);
        }
        async_wait0();      // next tile fully landed in LDS
        __syncthreads();    // all waves done reading cur, fills visible
    }

    #pragma unroll
    for (int tm = 0; tm < 2; tm++)
        #pragma unroll
        for (int tn = 0; tn < 2; tn++) {
            int n = n0 + wc * 32 + tn * 16 + idx;
            float bv = bias[n];
            #pragma unroll
            for (int j = 0; j < 8; j++) {
                int m = m0 + wr * 32 + tm * 16 + half * 8 + j;  // C: m = j + 8*half
                float v = acc[tm][tn][j] + bv;
                if (MODE == 1) v = 0.5f * v * (1.0f + erff(v * 0.70710678f));
                size_t off = (size_t)m * N + n;
                if (MODE == 2) v += R[off];
                C[off] = v;
            }
        }
}

// ---------------------------------------------------------------------------
// Feature attention: one block per (b,s). 128 threads -> (h = t>>5, fq = t&31).
// qkv rows: ((b*S+s)*F + f) * 768 : [q | k | v], head h at col h*64.
// Graph bias: gb_scale[h] * (graph==1 ? 0 : -5) + gb_offset[h].
// K/V staged with async b128 (32 KB each).
// ---------------------------------------------------------------------------
__global__ __launch_bounds__(128)
void feat_attn(const float* __restrict__ qkv, const float* __restrict__ graph,
               const float* __restrict__ gb_scale, const float* __restrict__ gb_offset,
               float* __restrict__ ctx)
{
    __shared__ __align__(16) float sK[F_ * D_];   // 32 KB
    __shared__ __align__(16) float sV[F_ * D_];   // 32 KB
    int bs  = blockIdx.x;
    int b   = bs >> 9;              // / S_
    int tid = threadIdx.x;
    for (int i = tid; i < (F_ * D_) / 4; i += 128) {  // 2048 16B segments
        int f = i >> 6;
        int d = (i & 63) * 4;
        size_t base = ((size_t)bs * F_ + f) * (3 * D_);
        async_ld128(lds_off32(&sK[f * D_ + d]), qkv + base + D_ + d);
        async_ld128(lds_off32(&sV[f * D_ + d]), qkv + base + 2 * D_ + d);
    }
    async_wait0();
    __syncthreads();
    int h = tid >> 5, fq = tid & 31;
    float q[DH_];
    {
        size_t base = ((size_t)bs * F_ + fq) * (3 * D_) + h * DH_;
        #pragma unroll
        for (int d = 0; d < DH_; d++) q[d] = qkv[base + d];
    }
    float gs = gb_scale[h], go = gb_offset[h];
    const float* grow = graph + (size_t)b * F_ * F_ + (size_t)fq * F_;
    float sc[F_];
    float mx = -1e30f;
    #pragma unroll
    for (int fk = 0; fk < F_; fk++) {
        const float* kp = &sK[fk * D_ + h * DH_];
        float dot = 0.f;
        #pragma unroll
        for (int d = 0; d < DH_; d++) dot += q[d] * kp[d];
        float bias = (grow[fk] == 1.0f) ? 0.0f : -5.0f;
        float s = dot * 0.125f + gs * bias + go;
        sc[fk] = s;
        mx = fmaxf(mx, s);
    }
    float l = 0.f;
    #pragma unroll
    for (int fk = 0; fk < F_; fk++) { float p = __expf(sc[fk] - mx); sc[fk] = p; l += p; }
    float acc[DH_];
    #pragma unroll
    for (int d = 0; d < DH_; d++) acc[d] = 0.f;
    #pragma unroll
    for (int fk = 0; fk < F_; fk++) {
        float p = sc[fk];
        const float* vp = &sV[fk * D_ + h * DH_];
        #pragma unroll
        for (int d = 0; d < DH_; d++) acc[d] += p * vp[d];
    }
    float inv = 1.0f / l;
    size_t obase = ((size_t)bs * F_ + fq) * D_ + h * DH_;
    #pragma unroll
    for (int d = 0; d < DH_; d++) ctx[obase + d] = acc[d] * inv;
}

// ---------------------------------------------------------------------------
// Sequence attention (train self-attn / test cross-attn), flash-style.
// grid = (B*F, H); 256 threads, one query per thread; online softmax over
// 32-key LDS tiles staged with async b128. Head col = col0 + h*64.
// ---------------------------------------------------------------------------
__global__ __launch_bounds__(256)
void seq_attn(const float* __restrict__ Q,  int q_stride, int q_col0,
              const float* __restrict__ Kp, int k_stride, int k_col0,
              const float* __restrict__ Vp, int v_stride, int v_col0,
              float* __restrict__ ctx, int nk)
{
    const int TK = 32;
    __shared__ __align__(16) float sK[TK * DH_];  // 8 KB
    __shared__ __align__(16) float sV[TK * DH_];  // 8 KB
    int bf = blockIdx.x, h = blockIdx.y;
    int tid = threadIdx.x;          // query index 0..255
    float q[DH_];
    {
        const float* qp = Q + ((size_t)bf * 256 + tid) * q_stride + q_col0 + h * DH_;
        #pragma unroll
        for (int d = 0; d < DH_; d++) q[d] = qp[d];
    }
    float m = -1e30f, l = 0.f;
    float acc[DH_];
    #pragma unroll
    for (int d = 0; d < DH_; d++) acc[d] = 0.f;

    for (int kt = 0; kt < nk; kt += TK) {
        #pragma unroll
        for (int rep = 0; rep < 2; rep++) {            // 512 16B segments
            int seg = tid + rep * 256;
            int kk = seg >> 4;
            int d  = (seg & 15) * 4;
            size_t krow = (size_t)bf * nk + kt + kk;
            async_ld128(lds_off32(&sK[kk * DH_ + d]),
                        Kp + krow * k_stride + k_col0 + h * DH_ + d);
            async_ld128(lds_off32(&sV[kk * DH_ + d]),
                        Vp + krow * v_stride + v_col0 + h * DH_ + d);
        }
        async_wait0();
        __syncthreads();
        float sc[TK];
        float tm = m;
        #pragma unroll
        for (int kk = 0; kk < TK; kk++) {
            const float* kp = &sK[kk * DH_];
            float dot = 0.f;
            #pragma unroll
            for (int d = 0; d < DH_; d++) dot += q[d] * kp[d];
            float s = dot * 0.125f;
            sc[kk] = s;
            tm = fmaxf(tm, s);
        }
        float corr = __expf(m - tm);
        l *= corr;
        #pragma unroll
        for (int d = 0; d < DH_; d++) acc[d] *= corr;
        #pragma unroll
        for (int kk = 0; kk < TK; kk++) {
            float p = __expf(sc[kk] - tm);
            l += p;
            const float* vp = &sV[kk * DH_];
            #pragma unroll
            for (int d = 0; d < DH_; d++) acc[d] += p * vp[d];
        }
        m = tm;
        __syncthreads();
    }
    float inv = 1.0f / l;
    size_t obase = ((size_t)bf * 256 + tid) * D_ + h * DH_;
    #pragma unroll
    for (int d = 0; d < DH_; d++) ctx[obase + d] = acc[d] * inv;
}

// ---------------------------------------------------------------------------
// Residual scatter-add: y rows are (b,f,s') contiguous; add into x1 at
// (b, s0+s', f, :).
// ---------------------------------------------------------------------------
__global__ __launch_bounds__(256)
void scatter_add(const float* __restrict__ y, float* __restrict__ x1, int s0)
{
    int r = blockIdx.x, d = threadIdx.x;
    int b   = r >> 13;
    int rem = r & 8191;
    int f   = rem >> 8;
    int sp  = rem & 255;
    size_t row = ((size_t)b * S_ + (size_t)(s0 + sp)) * F_ + f;
    x1[row * D_ + d] += y[(size_t)r * D_ + d];
}

// ---------------------------------------------------------------------------
extern "C" void kernel_launch(void* const* d_in, const int* in_sizes, int n_in,
                              void* d_out, int out_size, void* d_ws, size_t ws_size,
                              hipStream_t stream)
{
    (void)in_sizes; (void)n_in; (void)out_size; (void)ws_size;
    const float* x          = (const float*)d_in[0];
    const float* graph      = (const float*)d_in[1];
    const float* feat_in_w  = (const float*)d_in[4];
    const float* feat_in_b  = (const float*)d_in[5];
    const float* feat_out_w = (const float*)d_in[6];
    const float* feat_out_b = (const float*)d_in[7];
    const float* tr_in_w    = (const float*)d_in[8];
    const float* tr_in_b    = (const float*)d_in[9];
    const float* tr_out_w   = (const float*)d_in[10];
    const float* tr_out_b   = (const float*)d_in[11];
    const float* te_in_w    = (const float*)d_in[12];
    const float* te_in_b    = (const float*)d_in[13];
    const float* te_out_w   = (const float*)d_in[14];
    const float* te_out_b   = (const float*)d_in[15];
    const float* ln_feat_g  = (const float*)d_in[16];
    const float* ln_feat_b  = (const float*)d_in[17];
    const float* ln_tr_g    = (const float*)d_in[18];
    const float* ln_tr_b    = (const float*)d_in[19];
    const float* ln_te_g    = (const float*)d_in[20];
    const float* ln_te_b    = (const float*)d_in[21];
    const float* ln_ff_g    = (const float*)d_in[22];
    const float* ln_ff_b    = (const float*)d_in[23];
    const float* ff_w1      = (const float*)d_in[24];
    const float* ff_b1      = (const float*)d_in[25];
    const float* ff_w2      = (const float*)d_in[26];
    const float* ff_b2      = (const float*)d_in[27];
    const float* gb_scale   = (const float*)d_in[28];
    const float* gb_offset  = (const float*)d_in[29];
    float* out = (float*)d_out;

    const size_t NTOK  = (size_t)B_ * S_ * F_;   // 131072 rows
    const size_t NHALF = (size_t)B_ * F_ * 256;  // 65536 rows (train or test)

    float* ws   = (float*)d_ws;
    float* x1   = ws;                       // 33.5M floats
    float* bufA = x1   + NTOK * D_;         // 33.5M floats (LN out / qn+kvn)
    float* qkvb = bufA + NTOK * D_;         // 100.7M floats (qkv / q_te+kv_te / ff hidden)
    float* ctxb = qkvb + NTOK * 3 * D_;     // 33.5M floats (ctx / ctx+y)
    float* ybuf = ctxb + NHALF * D_;
    float* qn   = bufA;
    float* kvn  = bufA + NHALF * D_;
    float* qte  = qkvb;
    float* kvte = qkvb + NHALF * D_;
    float* hbuf = qkvb;

    dim3 blk256(256), blk128(128);

    // ---- Stage 1: feature attention over F with graph bias -------------
    ln_rows<<<(int)NTOK, blk256, 0, stream>>>(x, bufA, ln_feat_g, ln_feat_b, 0, 0);
    gemm_wmma<0><<<dim3(768/64, (int)(NTOK/128)), blk256, 0, stream>>>(
        bufA, feat_in_w, feat_in_b, nullptr, qkvb, (int)NTOK, 768, 256);
    feat_attn<<<B_ * S_, blk128, 0, stream>>>(qkvb, graph, gb_scale, gb_offset, ctxb);
    gemm_wmma<2><<<dim3(256/64, (int)(NTOK/128)), blk256, 0, stream>>>(
        ctxb, feat_out_w, feat_out_b, x, x1, (int)NTOK, 256, 256);

    // ---- Stage 2: train self-attention over s ---------------------------
    ln_rows<<<(int)NHALF, blk256, 0, stream>>>(x1, bufA, ln_tr_g, ln_tr_b, 1, 0);
    gemm_wmma<0><<<dim3(768/64, (int)(NHALF/128)), blk256, 0, stream>>>(
        bufA, tr_in_w, tr_in_b, nullptr, qkvb, (int)NHALF, 768, 256);
    seq_attn<<<dim3(B_ * F_, H_), blk256, 0, stream>>>(
        qkvb, 768, 0, qkvb, 768, 256, qkvb, 768, 512, ctxb, 256);
    gemm_wmma<0><<<dim3(256/64, (int)(NHALF/128)), blk256, 0, stream>>>(
        ctxb, tr_out_w, tr_out_b, nullptr, ybuf, (int)NHALF, 256, 256);
    scatter_add<<<(int)NHALF, blk256, 0, stream>>>(ybuf, x1, 0);

    // ---- Stage 3: test cross-attention (q=test, kv=updated train) ------
    ln_rows<<<(int)NHALF, blk256, 0, stream>>>(x1, qn,  ln_te_g, ln_te_b, 1, 256);
    ln_rows<<<(int)NHALF, blk256, 0, stream>>>(x1, kvn, ln_te_g, ln_te_b, 1, 0);
    gemm_wmma<0><<<dim3(256/64, (int)(NHALF/128)), blk256, 0, stream>>>(
        qn, te_in_w, te_in_b, nullptr, qte, (int)NHALF, 256, 256);
    gemm_wmma<0><<<dim3(512/64, (int)(NHALF/128)), blk256, 0, stream>>>(
        kvn, te_in_w + 256 * 256, te_in_b + 256, nullptr, kvte, (int)NHALF, 512, 256);
    seq_attn<<<dim3(B_ * F_, H_), blk256, 0, stream>>>(
        qte, 256, 0, kvte, 512, 0, kvte, 512, 256, ctxb, 256);
    gemm_wmma<0><<<dim3(256/64, (int)(NHALF/128)), blk256, 0, stream>>>(
        ctxb, te_out_w, te_out_b, nullptr, ybuf, (int)NHALF, 256, 256);
    scatter_add<<<(int)NHALF, blk256, 0, stream>>>(ybuf, x1, 256);

    // ---- Stage 4: FFN with exact GELU, residual to d_out ----------------
    ln_rows<<<(int)NTOK, blk256, 0, stream>>>(x1, bufA, ln_ff_g, ln_ff_b, 0, 0);
    gemm_wmma<1><<<dim3(512/64, (int)(NTOK/128)), blk256, 0, stream>>>(
        bufA, ff_w1, ff_b1, nullptr, hbuf, (int)NTOK, 512, 256);
    gemm_wmma<2><<<dim3(256/64, (int)(NTOK/128)), blk256, 0, stream>>>(
        hbuf, ff_w2, ff_b2, x1, out, (int)NTOK, 256, 512);
}